// ImVoxelViewTransform_10264971837633
// MI455X (gfx1250) — compile-verified
//
#include <hip/hip_runtime.h>
#include <math.h>

// ---- problem constants (from reference) ----
#define NXv   128
#define NYv   128
#define NZv   8
#define NVOX  (NXv*NYv*NZv)      // 131072
#define NVIEW 6
#define CCH   64
#define HFm   116
#define WFm   200
#define IMGW  1600.0f
#define IMGH  928.0f
#define SCL   0.125f             // WF/PAD_W == HF/PAD_H == 0.125

typedef __attribute__((ext_vector_type(2))) float v2f;
typedef __attribute__((ext_vector_type(8))) float v8f;

// -------------------------------------------------------------------------
// Stage 1: [V,C,H,W] -> [V,H,W,C]   (channels-last so a bilinear corner's 64
// channels are one contiguous 256B read instead of 64 strided cache lines)
// -------------------------------------------------------------------------
__global__ __launch_bounds__(256)
void transpose_feat(const float* __restrict__ in, float* __restrict__ outT) {
    const long long total = (long long)NVIEW * CCH * HFm * WFm;
    for (long long idx = (long long)blockIdx.x * blockDim.x + threadIdx.x; idx < total;
         idx += (long long)gridDim.x * blockDim.x) {
        int c = (int)(idx & (CCH - 1));
        long long t = idx >> 6;
        int w = (int)(t % WFm); t /= WFm;
        int h = (int)(t % HFm);
        int v = (int)(t / HFm);
        // write coalesced (c innermost); reads reuse lines via L1/L2
        outT[idx] = in[(((long long)v * CCH + c) * HFm + h) * WFm + w];
    }
}

// -------------------------------------------------------------------------
// Stage 2: WMMA projection. D = A(16x4) * B(4x16), fp32.
//   A rows m = (view, row) pairs; B cols = 16 points (homogeneous coords).
//   C/D layout: lane n (0-15) holds cols, VGPR r holds row m=r (lanes 0-15)
//   or m=r+8 (lanes 16-31)  -> each lane gets rows 0-3 of two views for ONE
//   point, so u/v/depth/valid computed entirely lane-locally.
// -------------------------------------------------------------------------
__device__ __forceinline__ void emit_proj(float r0, float r1, float r2, int v, int p,
                                          float4* __restrict__ pxpyv) {
    float ds  = fabsf(r2) > 1e-6f ? r2 : 1e-6f;
    float inv = 1.0f / ds;                 // one precise divide, two multiplies
    float u = r0 * inv;
    float w = r1 * inv;
    bool valid = (r2 > 0.f) && (u > 0.f) && (u < IMGW) && (w > 0.f) && (w < IMGH);
    float4 q;
    q.x = u * SCL - 0.5f;
    q.y = w * SCL - 0.5f;
    q.z = valid ? 1.0f : 0.0f;
    q.w = 0.0f;
    pxpyv[(long long)v * NVOX + p] = q;
}

__global__ __launch_bounds__(256)
void project_wmma(const float* __restrict__ points,
                  const float* __restrict__ proj,   // [6][4][4] row-major
                  float4* __restrict__ pxpyv) {
    const int lane = threadIdx.x & 31;
    const int wave = threadIdx.x >> 5;
    const int base = (blockIdx.x * 8 + wave) * 16;   // 16 points per wave

    const int m    = lane & 15;      // A-matrix row owned by this lane pair
    const int half = lane >> 4;      // 0 -> K={0,1}, 1 -> K={2,3}
    const int k0   = half * 2;
    const int row  = m & 3;
    const int v1   = m >> 2;               // views 0..3
    const int v2   = 4 + ((m >> 2) & 1);   // views 4,5,4,5

    // A operands (proj-matrix rows), per documented 32-bit 16x4 A layout
    v2f a1, a2, b;
    a1[0] = proj[v1 * 16 + row * 4 + k0];
    a1[1] = proj[v1 * 16 + row * 4 + k0 + 1];
    a2[0] = proj[v2 * 16 + row * 4 + k0];
    a2[1] = proj[v2 * 16 + row * 4 + k0 + 1];

    // B operand: column = point, rows K = (x,y,z,1)
    const int p = base + m;
    float bx = points[3 * p + 0];
    float by = points[3 * p + 1];
    float bz = points[3 * p + 2];
    b[0] = half ? bz : bx;
    b[1] = half ? 1.0f : by;

    v8f cz = {0.f, 0.f, 0.f, 0.f, 0.f, 0.f, 0.f, 0.f};
    // 8 args: (neg_a, A, neg_b, B, c_mod, C, reuse_a, reuse_b)
    v8f d1 = __builtin_amdgcn_wmma_f32_16x16x4_f32(false, a1, false, b, (short)0, cz, false, false);
    v8f d2 = __builtin_amdgcn_wmma_f32_16x16x4_f32(false, a2, false, b, (short)0, cz, false, false);

    if (half == 0) {
        const int pp = base + lane;          // this lane's point
        emit_proj(d1[0], d1[1], d1[2], 0, pp, pxpyv);
        emit_proj(d1[4], d1[5], d1[6], 1, pp, pxpyv);
        emit_proj(d2[0], d2[1], d2[2], 4, pp, pxpyv);
        emit_proj(d2[4], d2[5], d2[6], 5, pp, pxpyv);
    } else {
        const int pp = base + lane - 16;
        emit_proj(d1[0], d1[1], d1[2], 2, pp, pxpyv);  // m=8..11  -> view 2
        emit_proj(d1[4], d1[5], d1[6], 3, pp, pxpyv);  // m=12..15 -> view 3
        // d2 here duplicates views 4/5 of the same points -> skip
    }
}

// -------------------------------------------------------------------------
// Stage 3: gather + bilinear + masked average; 64 f32 accumulators/thread.
// -------------------------------------------------------------------------
__device__ __forceinline__ void fma4(float4& a, float w, const float4& f) {
    a.x = fmaf(w, f.x, a.x);
    a.y = fmaf(w, f.y, a.y);
    a.z = fmaf(w, f.z, a.z);
    a.w = fmaf(w, f.w, a.w);
}

__global__ __launch_bounds__(256)
void gather_avg(const float4* __restrict__ featT,   // [V,H,W,C/4]
                const float4* __restrict__ pxpyv,   // [V,N] (px,py,valid,0)
                float* __restrict__ out) {
    const int q = blockIdx.x * blockDim.x + threadIdx.x;   // 0..131071
    const int x = q >> 10;
    const int r = q & 1023;
    const int n = ((r & 7) << 14) + ((r >> 3) << 7) + x;   // voxel flat index

    // issue all six projection-record loads up front (overlap latency)
    float4 pv[NVIEW];
#pragma unroll
    for (int v = 0; v < NVIEW; ++v) pv[v] = pxpyv[v * NVOX + n];

    float4 acc[16];
#pragma unroll
    for (int i = 0; i < 16; ++i) acc[i] = make_float4(0.f, 0.f, 0.f, 0.f);
    float cnt = 0.f;

#pragma unroll
    for (int v = 0; v < NVIEW; ++v) {
        if (pv[v].z > 0.f) {
            cnt += 1.f;
            float x0f = floorf(pv[v].x), y0f = floorf(pv[v].y);
            int   xi = (int)x0f,         yi = (int)y0f;
            float wx1 = pv[v].x - x0f,   wy1 = pv[v].y - y0f;
            float wx0 = 1.f - wx1,       wy0 = 1.f - wy1;

            float ww[4] = { wx0 * wy0, wx1 * wy0, wx0 * wy1, wx1 * wy1 };
            int   cx[4] = { xi, xi + 1, xi,     xi + 1 };
            int   cy[4] = { yi, yi,     yi + 1, yi + 1 };
            int   cb[4];
#pragma unroll
            for (int k = 0; k < 4; ++k) {
                bool inb = (cx[k] >= 0) && (cx[k] < WFm) && (cy[k] >= 0) && (cy[k] < HFm);
                int xc = min(max(cx[k], 0), WFm - 1);
                int yc = min(max(cy[k], 0), HFm - 1);
                ww[k] = inb ? ww[k] : 0.f;                  // zero-padding mode
                cb[k] = ((v * HFm + yc) * WFm + xc) * (CCH / 4);
            }
#pragma unroll
            for (int cc = 0; cc < 16; ++cc) {
                float4 f0 = featT[cb[0] + cc];
                float4 f1 = featT[cb[1] + cc];
                float4 f2 = featT[cb[2] + cc];
                float4 f3 = featT[cb[3] + cc];
                fma4(acc[cc], ww[0], f0);
                fma4(acc[cc], ww[1], f1);
                fma4(acc[cc], ww[2], f2);
                fma4(acc[cc], ww[3], f3);
            }
        }
    }

    float inv = cnt > 0.f ? 1.f / cnt : 0.f;  // cnt==0 -> acc already zero
    // out[c,x,y,z] = vol[c, n]; address = c*NVOX + x*1024 + r == c*NVOX + q
#pragma unroll
    for (int cc = 0; cc < 16; ++cc) {
        out[(4 * cc + 0) * NVOX + q] = acc[cc].x * inv;
        out[(4 * cc + 1) * NVOX + q] = acc[cc].y * inv;
        out[(4 * cc + 2) * NVOX + q] = acc[cc].z * inv;
        out[(4 * cc + 3) * NVOX + q] = acc[cc].w * inv;
    }
}

// -------------------------------------------------------------------------
// Fallback (ws too small): monolithic VALU path on original layout.
// -------------------------------------------------------------------------
__global__ __launch_bounds__(256)
void fallback_all(const float* __restrict__ xfov,
                  const float* __restrict__ points,
                  const float* __restrict__ proj,
                  float* __restrict__ out) {
    const int q = blockIdx.x * blockDim.x + threadIdx.x;
    const int x = q >> 10;
    const int r = q & 1023;
    const int n = ((r & 7) << 14) + ((r >> 3) << 7) + x;

    float px = points[3 * n], py = points[3 * n + 1], pz = points[3 * n + 2];
    float acc[CCH];
#pragma unroll
    for (int c = 0; c < CCH; ++c) acc[c] = 0.f;
    float cnt = 0.f;

    for (int v = 0; v < NVIEW; ++v) {
        const float* M = proj + v * 16;
        float r0 = M[0] * px + M[1] * py + M[2] * pz + M[3];
        float r1 = M[4] * px + M[5] * py + M[6] * pz + M[7];
        float r2 = M[8] * px + M[9] * py + M[10] * pz + M[11];
        float ds  = fabsf(r2) > 1e-6f ? r2 : 1e-6f;
        float inv = 1.0f / ds;
        float u = r0 * inv, w = r1 * inv;
        bool valid = (r2 > 0.f) && (u > 0.f) && (u < IMGW) && (w > 0.f) && (w < IMGH);
        if (!valid) continue;
        cnt += 1.f;
        float sx = u * SCL - 0.5f, sy = w * SCL - 0.5f;
        float x0f = floorf(sx), y0f = floorf(sy);
        int   xi = (int)x0f,    yi = (int)y0f;
        float wx1 = sx - x0f,   wy1 = sy - y0f;
        float wx0 = 1.f - wx1,  wy0 = 1.f - wy1;
        float ww[4] = { wx0 * wy0, wx1 * wy0, wx0 * wy1, wx1 * wy1 };
        int   cx[4] = { xi, xi + 1, xi,     xi + 1 };
        int   cy[4] = { yi, yi,     yi + 1, yi + 1 };
        int   cb[4];
#pragma unroll
        for (int k = 0; k < 4; ++k) {
            bool inb = (cx[k] >= 0) && (cx[k] < WFm) && (cy[k] >= 0) && (cy[k] < HFm);
            int xc = min(max(cx[k], 0), WFm - 1);
            int yc = min(max(cy[k], 0), HFm - 1);
            ww[k] = inb ? ww[k] : 0.f;
            cb[k] = yc * WFm + xc;
        }
        const float* F = xfov + (long long)v * CCH * HFm * WFm;
        for (int c = 0; c < CCH; ++c) {
            const float* Fc = F + (long long)c * HFm * WFm;
            acc[c] += ww[0] * Fc[cb[0]] + ww[1] * Fc[cb[1]] +
                      ww[2] * Fc[cb[2]] + ww[3] * Fc[cb[3]];
        }
    }
    float inv = cnt > 0.f ? 1.f / cnt : 0.f;
#pragma unroll
    for (int c = 0; c < CCH; ++c) out[c * NVOX + q] = acc[c] * inv;
}

// -------------------------------------------------------------------------
extern "C" void kernel_launch(void* const* d_in, const int* in_sizes, int n_in,
                              void* d_out, int out_size, void* d_ws, size_t ws_size,
                              hipStream_t stream) {
    const float* x_fov  = (const float*)d_in[0];   // [1,6,64,116,200]
    const float* points = (const float*)d_in[1];   // [131072,3]
    const float* proj   = (const float*)d_in[2];   // [6,4,4]
    float* out = (float*)d_out;                    // [1,64,128,128,8]

    const size_t featElems = (size_t)NVIEW * HFm * WFm * CCH;       // 8,908,800
    const size_t need = featElems * sizeof(float)                    // transposed feat
                      + (size_t)NVIEW * NVOX * sizeof(float4);       // (px,py,valid,0)

    if (ws_size >= need) {
        float*  feat_t = (float*)d_ws;
        float4* pxpyv  = (float4*)(feat_t + featElems);

        {   // stage 1: channels-last transpose
            long long total = (long long)NVIEW * CCH * HFm * WFm;
            int blocks = (int)((total + 255) / 256);
            transpose_feat<<<blocks, 256, 0, stream>>>(x_fov, feat_t);
        }
        {   // stage 2: WMMA projection (16 points/wave, 8 waves/block)
            project_wmma<<<NVOX / 128, 256, 0, stream>>>(points, proj, pxpyv);
        }
        {   // stage 3: gather + average
            gather_avg<<<NVOX / 256, 256, 0, stream>>>((const float4*)feat_t, pxpyv, out);
        }
    } else {
        fallback_all<<<NVOX / 256, 256, 0, stream>>>(x_fov, points, proj, out);
    }
}